// MonarchMixerLayer_5111011082511
// MI455X (gfx1250) — compile-verified
//
#include <hip/hip_runtime.h>

// Monarch Mixer layer for MI455X (gfx1250), fp32 via V_WMMA_F32_16X16X4_F32.
// out[k*s+n] = sum_q R[n,k,q] * sum_p L[q,n,p] * x[p*s+q]  per row.
// Memory-bound (~150MB min traffic @ 23.3 TB/s); intermediates ping-pong in the
// 192MB L2. GEMMs arranged as C[j][d] = W[j][k]*X[k][d] so WMMA N = contiguous d.
// LDS tiles are K-pair interleaved: each WMMA B fragment (rows ka, ka+1 with
// ka = k0+2h per the ISA A/B layout) is one aligned ds_load_b64 straight into
// an even VGPR pair -- no v_mov assembly before v_wmma.

typedef float v2f __attribute__((ext_vector_type(2)));
typedef float v8f __attribute__((ext_vector_type(8)));

#define NB    4
#define NSEQ  4096   // N = 64*64
#define ND    1024   // D = 32*32
#define SPAIR 160    // seq tile pair-row stride (== 32 mod 64 -> conflict-free b64)
#define RSF   1032   // feat X row stride (even -> 8B-aligned b64)

static __device__ __forceinline__ v8f wmma4(v2f a, v2f b, v8f c) {
  // (neg_a, A, neg_b, B, c_mod, C, reuse_a, reuse_b)
  return __builtin_amdgcn_wmma_f32_16x16x4_f32(false, a, false, b, (short)0, c, false, false);
}

// p-pair permutation of d for the feature X buffer: d = p*32+q  ->
// (p>>1)*64 + q*2 + (p&1). Stage1 B-frags and stage2 D-stores use it natively.
static __device__ __forceinline__ int permidx(int d) {
  return ((d >> 6) << 6) + ((d & 31) << 1) + ((d >> 5) & 1);
}

// ---------------------------------------------------------------------------
// Sequence-mixing stage 1:  T[b][n][q][d] = sum_p W[q][n][p] * in[b][p*64+q][d]
// grid = (q=64, dtile=16, b=4), 256 threads (8 waves).
// ---------------------------------------------------------------------------
__global__ __launch_bounds__(256) void k_seq_s1(const float* __restrict__ in,
                                                const float* __restrict__ W,
                                                float* __restrict__ T) {
  const int q = blockIdx.x, dt = blockIdx.y, b = blockIdx.z;
  const int d0 = dt * 64;
  __shared__ float Xs[32 * SPAIR];                    // (p,d) -> (p>>1)*SPAIR + d*2 + (p&1)
  const int tid = threadIdx.x;
  for (int i = tid; i < 64 * 64; i += 256) {          // coalesced 64-float rows in
    int p = i >> 6, dl = i & 63;
    Xs[(p >> 1) * SPAIR + dl * 2 + (p & 1)] =
        in[((size_t)(b * NSEQ + p * 64 + q)) * ND + d0 + dl];
  }
  __syncthreads();

  const int wave = tid >> 5, lane = tid & 31;
  const int l16 = lane & 15, h = lane >> 4;
  const int mt  = wave & 3;                           // n-tile (M)
  const int nt0 = (wave >> 2) * 2;                    // d-tile pair (N)
  const float* Wq = W + (size_t)q * 4096;             // W[q][n][p]
  const int arow = (mt * 16 + l16) * 64;
  const int dA = (nt0 * 16 + l16) * 2, dB = dA + 32;

  v8f acc0 = {}, acc1 = {};
  for (int k0 = 0; k0 < 64; k0 += 4) {
    int pp = (k0 >> 1) + h;                           // pair row: rows (2pp, 2pp+1)
    v2f a = *(const v2f*)(Wq + arow + k0 + 2 * h);    // A: K = k0+2h, k0+2h+1
    const float* bp = &Xs[pp * SPAIR];
    v2f b0 = *(const v2f*)(bp + dA);                  // single ds_load_b64 each
    v2f b1 = *(const v2f*)(bp + dB);
    acc0 = wmma4(a, b0, acc0);
    acc1 = wmma4(a, b1, acc1);
  }
#pragma unroll
  for (int v = 0; v < 8; ++v) {                       // D: M = v+8h, N = l16
    int n = mt * 16 + v + 8 * h;
    size_t base = ((size_t)(b * 64 + n) * 64 + q) * ND + d0;
    T[base + nt0 * 16 + l16]       = acc0[v];
    T[base + (nt0 + 1) * 16 + l16] = acc1[v];
  }
}

// ---------------------------------------------------------------------------
// Sequence-mixing stage 2:
//   out[b][k*64+n][d] = (sum_q W[n][k][q] * T[b][n][q][d]) * (nkT ? nkT[k*64+n][d] : 1)
// grid = (n=64, dtile=16, b=4).
// ---------------------------------------------------------------------------
__global__ __launch_bounds__(256) void k_seq_s2(const float* __restrict__ T,
                                                const float* __restrict__ W,
                                                const float* __restrict__ nkT,
                                                float* __restrict__ out) {
  const int n = blockIdx.x, dt = blockIdx.y, b = blockIdx.z;
  const int d0 = dt * 64;
  __shared__ float Xs[32 * SPAIR];                    // (q,d) pair-interleaved
  const int tid = threadIdx.x;
  for (int i = tid; i < 64 * 64; i += 256) {
    int qq = i >> 6, dl = i & 63;
    Xs[(qq >> 1) * SPAIR + dl * 2 + (qq & 1)] =
        T[((size_t)(b * 64 + n) * 64 + qq) * ND + d0 + dl];
  }
  __syncthreads();

  const int wave = tid >> 5, lane = tid & 31;
  const int l16 = lane & 15, h = lane >> 4;
  const int mt  = wave & 3;                           // k-tile
  const int nt0 = (wave >> 2) * 2;
  const float* Wn = W + (size_t)n * 4096;             // W[n][k][q]
  const int arow = (mt * 16 + l16) * 64;
  const int dA = (nt0 * 16 + l16) * 2, dB = dA + 32;

  v8f acc0 = {}, acc1 = {};
  for (int k0 = 0; k0 < 64; k0 += 4) {
    int pp = (k0 >> 1) + h;
    v2f a = *(const v2f*)(Wn + arow + k0 + 2 * h);
    const float* bp = &Xs[pp * SPAIR];
    v2f b0 = *(const v2f*)(bp + dA);
    v2f b1 = *(const v2f*)(bp + dB);
    acc0 = wmma4(a, b0, acc0);
    acc1 = wmma4(a, b1, acc1);
  }
#pragma unroll
  for (int v = 0; v < 8; ++v) {
    int k = mt * 16 + v + 8 * h;
    size_t i = (size_t)(k * 64 + n);
    size_t obase = ((size_t)b * NSEQ + i) * ND + d0;
    float f0 = acc0[v], f1 = acc1[v];
    if (nkT) {                                        // coalesced 64B gathers
      f0 *= nkT[i * ND + d0 + nt0 * 16 + l16];
      f1 *= nkT[i * ND + d0 + (nt0 + 1) * 16 + l16];
    }
    out[obase + nt0 * 16 + l16]       = f0;
    out[obase + (nt0 + 1) * 16 + l16] = f1;
  }
}

// ---------------------------------------------------------------------------
// n_kernel transpose: nkT[i][d] = nk[d][i]   (1024 x 4096 -> 4096 x 1024)
// ---------------------------------------------------------------------------
__global__ __launch_bounds__(256) void k_transpose(const float* __restrict__ src,
                                                   float* __restrict__ dst) {
  __shared__ float tile[32][33];
  const int c0 = blockIdx.x * 32, r0 = blockIdx.y * 32;
  const int tx = threadIdx.x & 31, ty = threadIdx.x >> 5;
#pragma unroll
  for (int k = 0; k < 4; ++k) {
    int r = ty + 8 * k;
    tile[r][tx] = src[(size_t)(r0 + r) * NSEQ + c0 + tx];
  }
  __syncthreads();
#pragma unroll
  for (int k = 0; k < 4; ++k) {
    int r = ty + 8 * k;
    dst[(size_t)(c0 + r) * ND + r0 + tx] = tile[tx][r];
  }
}

// ---------------------------------------------------------------------------
// Feature-mixing fused monarch (s=32): 16 rows staged in LDS, both stages +
// epilogue fused.  mode 0: out = relu(dk[d]*monarch(in));
// mode 1: out = LayerNorm(monarch(in) + resid).
// Dynamic LDS: X[16][RSF] (p-pair permuted d) + T[32][16 qq][16 r][2] ~ 128.5KB.
// ---------------------------------------------------------------------------
__global__ __launch_bounds__(256) void k_feat(const float* __restrict__ in,
                                              const float* __restrict__ L,
                                              const float* __restrict__ R,
                                              const float* __restrict__ dk,
                                              const float* __restrict__ resid,
                                              const float* __restrict__ lnw,
                                              const float* __restrict__ lnb,
                                              float* __restrict__ out,
                                              int mode) {
  extern __shared__ float lds[];
  float* X = lds;                                     // [r][RSF], d permuted
  float* T = lds + 16 * RSF;                          // (n,q,r) -> n*512 + (q>>1)*32 + r*2 + (q&1)
  const int row0 = blockIdx.x * 16;
  const int tid  = threadIdx.x;

  for (int i = tid; i < 16 * ND; i += 256) {          // coalesced row loads
    int r = i >> 10, d = i & 1023;
    X[r * RSF + permidx(d)] = in[(size_t)(row0 + r) * ND + d];
  }
  __syncthreads();

  const int wave = tid >> 5, lane = tid & 31;
  const int l16 = lane & 15, h = lane >> 4;

  // stage 1: T[n][q][r] = sum_p L[q][n][p] * X[r][p*32+q]
  for (int j = 0; j < 4; ++j) {
    int q = wave + 8 * j;
    const float* Lq = L + (size_t)q * 1024;
#pragma unroll
    for (int mt = 0; mt < 2; ++mt) {
      v8f acc = {};
      const int arow = (mt * 16 + l16) * 32;
      for (int k0 = 0; k0 < 32; k0 += 4) {
        int pp = (k0 >> 1) + h;                       // rows (2pp, 2pp+1) of p
        v2f a  = *(const v2f*)(Lq + arow + k0 + 2 * h);
        v2f bb = *(const v2f*)(&X[l16 * RSF + pp * 64 + q * 2]);
        acc = wmma4(a, bb, acc);
      }
#pragma unroll
      for (int v = 0; v < 8; ++v) {
        int nn = mt * 16 + v + 8 * h;
        T[nn * 512 + (q >> 1) * 32 + l16 * 2 + (q & 1)] = acc[v];
      }
    }
  }
  __syncthreads();

  // stage 2: X[r][k*32+n] = sum_q R[n][k][q] * T[n][q][r]
  for (int j = 0; j < 4; ++j) {
    int n = wave + 8 * j;
    const float* Rn = R + (size_t)n * 1024;
#pragma unroll
    for (int mt = 0; mt < 2; ++mt) {
      v8f acc = {};
      const int arow = (mt * 16 + l16) * 32;
      for (int k0 = 0; k0 < 32; k0 += 4) {
        int pp = (k0 >> 1) + h;                       // q-pairs (2pp, 2pp+1)
        v2f a  = *(const v2f*)(Rn + arow + k0 + 2 * h);
        v2f bb = *(const v2f*)(&T[n * 512 + pp * 32 + l16 * 2]);  // conflict-free b64
        acc = wmma4(a, bb, acc);
      }
#pragma unroll
      for (int v = 0; v < 8; ++v) {
        int k = mt * 16 + v + 8 * h;
        X[l16 * RSF + (k >> 1) * 64 + n * 2 + (k & 1)] = acc[v];
      }
    }
  }
  __syncthreads();

  if (mode == 0) {                                    // d_kernel * , relu
    for (int i = tid; i < 16 * ND; i += 256) {
      int r = i >> 10, d = i & 1023;
      float v = X[r * RSF + permidx(d)] * dk[d];
      out[(size_t)(row0 + r) * ND + d] = v > 0.f ? v : 0.f;
    }
  } else {                                            // residual + LayerNorm
    for (int i = tid; i < 16 * ND; i += 256) {
      int r = i >> 10, d = i & 1023;
      X[r * RSF + permidx(d)] += resid[(size_t)(row0 + r) * ND + d];
    }
    __syncthreads();
    {                                                 // group of 16 lanes per row
      int r = tid >> 4, l = tid & 15;
      float s = 0.f, ss = 0.f;
      for (int d = l; d < ND; d += 16) {
        float v = X[r * RSF + permidx(d)];
        s += v; ss += v * v;
      }
#pragma unroll
      for (int off = 8; off > 0; off >>= 1) {         // stays within 16-lane group
        s  += __shfl_xor(s,  off, 32);
        ss += __shfl_xor(ss, off, 32);
      }
      if (l == 0) {
        float mu  = s * (1.f / ND);
        float var = ss * (1.f / ND) - mu * mu;
        T[r * 2]     = mu;                            // T region free now
        T[r * 2 + 1] = rsqrtf(var + 1e-5f);
      }
    }
    __syncthreads();
    for (int i = tid; i < 16 * ND; i += 256) {
      int r = i >> 10, d = i & 1023;
      float mu = T[r * 2], rs = T[r * 2 + 1];
      out[(size_t)(row0 + r) * ND + d] =
          (X[r * RSF + permidx(d)] - mu) * rs * lnw[d] + lnb[d];
    }
  }
}

// ---------------------------------------------------------------------------
extern "C" void kernel_launch(void* const* d_in, const int* in_sizes, int n_in,
                              void* d_out, int out_size, void* d_ws, size_t ws_size,
                              hipStream_t stream) {
  (void)in_sizes; (void)n_in; (void)out_size; (void)ws_size;
  const float* x   = (const float*)d_in[0];
  const float* m1L = (const float*)d_in[1];
  const float* m1R = (const float*)d_in[2];
  const float* m2L = (const float*)d_in[3];
  const float* m2R = (const float*)d_in[4];
  const float* m3L = (const float*)d_in[5];
  const float* m3R = (const float*)d_in[6];
  const float* m4L = (const float*)d_in[7];
  const float* m4R = (const float*)d_in[8];
  const float* nk  = (const float*)d_in[9];
  const float* dk  = (const float*)d_in[10];
  const float* lnw = (const float*)d_in[11];
  const float* lnb = (const float*)d_in[12];
  float* out = (float*)d_out;

  const size_t NEL = (size_t)NB * NSEQ * ND;          // 16M floats = 64MB
  float* bufA = (float*)d_ws;
  float* bufB = bufA + NEL;
  float* nkT  = bufB + NEL;                           // 16MB

  k_transpose<<<dim3(NSEQ / 32, ND / 32), 256, 0, stream>>>(nk, nkT);

  dim3 gseq(64, 16, NB);
  k_seq_s1<<<gseq, 256, 0, stream>>>(x,    m1L, bufA);          // T1
  k_seq_s2<<<gseq, 256, 0, stream>>>(bufA, m1R, nkT, bufB);     // nk * monarch1
  k_seq_s1<<<gseq, 256, 0, stream>>>(bufB, m2L, bufA);          // T2
  k_seq_s2<<<gseq, 256, 0, stream>>>(bufA, m2R, nullptr, bufB); // x_tilde

  const size_t featLds = (size_t)(16 * RSF + 32 * 512) * sizeof(float);
  k_feat<<<NB * NSEQ / 16, 256, featLds, stream>>>(bufB, m3L, m3R, dk,
                                                   nullptr, nullptr, nullptr,
                                                   bufA, 0);    // y3
  k_feat<<<NB * NSEQ / 16, 256, featLds, stream>>>(bufA, m4L, m4R, nullptr,
                                                   bufB, lnw, lnb,
                                                   out, 1);     // LN(y4 + x_tilde)
}